// InputLayer_57337813401914
// MI455X (gfx1250) — compile-verified
//
#include <hip/hip_runtime.h>
#include <hip/hip_bf16.h>

// out[row, v] = sin(x[row] * freq[v]),  row in [0, 64*4096), v in [0, 256)
// freq[v] = 10^(min(v,127)/127 * 4)    (upper half clamps to 10^4)
//
// MI455X roofline: 256 MiB NT stores / 23.3 TB/s ~= 11.5 us -> store-bound.
// Fast path (V=256 hardcoded): per-thread register freqs (v_exp_f32 once),
// v_sin_f32 on the TRANS pipe, one b128 load of 4 rows of x per macro-iter
// (wave-uniform -> L0 broadcast), 4x global_store_b128 th:NT.

typedef __attribute__((ext_vector_type(4))) float v4f;

#define ONE_OVER_2PI 0.15915494309189535f
#define LOG2_10      3.321928094887362f

#define V256   256
#define BLOCK  256
#define VLANES (V256 / 4)            // 64 threads per row
#define RLANES (BLOCK / VLANES)      // 4 row-groups per block
#define HALFM1 (V256 / 2 - 1)        // 127

__global__ __launch_bounds__(BLOCK) void sin_posenc256(
    const float* __restrict__ x,
    float* __restrict__ out,
    int n_rows,
    int rows_per_block)
{
    const int vlane = (int)threadIdx.x & (VLANES - 1);
    const int rlane = (int)threadIdx.x / VLANES;     // shift, wave-uniform
    const int v0    = vlane << 2;

    // Per-thread frequencies, computed ONCE, pre-folded with 1/(2*pi):
    // inner loop is just v_mul_f32 + v_sin_f32 per output element.
    const float c = (4.0f * LOG2_10) / (float)HALFM1;
    float g[4];
#pragma unroll
    for (int k = 0; k < 4; ++k) {
        int j = v0 + k;
        if (j > HALFM1) j = HALFM1;                  // clamp: upper half -> 10^4
        g[k] = __builtin_exp2f(c * (float)j) * ONE_OVER_2PI;  // v_exp_f32
    }

    const int block_base = (int)blockIdx.x * rows_per_block;
    int row_stop = block_base + rows_per_block;
    if (row_stop > n_rows) row_stop = n_rows;

    // Each thread owns 4 consecutive rows per macro-iteration:
    // 1x global_load_b128 (wave-uniform -> broadcast) + 4x b128 NT stores.
    for (int r0 = block_base + rlane * 4; r0 + 3 < row_stop; r0 += RLANES * 4) {
        const v4f xv = *(const v4f*)(x + r0);

#pragma unroll
        for (int rr = 0; rr < 4; ++rr) {
            const float xs = ((const float*)&xv)[rr];
            v4f r;
            r.x = __builtin_amdgcn_sinf(xs * g[0]);  // v_sin_f32 (TRANS)
            r.y = __builtin_amdgcn_sinf(xs * g[1]);
            r.z = __builtin_amdgcn_sinf(xs * g[2]);
            r.w = __builtin_amdgcn_sinf(xs * g[3]);

            v4f* dst = (v4f*)(out + (size_t)(r0 + rr) * (size_t)V256 + (size_t)v0);
            __builtin_nontemporal_store(r, dst);     // global_store_b128 th:NT
        }
    }
}

// Minimal scalar fallback for unexpected V (never launched for the reference
// shape): 1 element/thread, row = blockIdx.y, v_rcp instead of IEEE div.
__global__ __launch_bounds__(256) void sin_posenc_any(
    const float* __restrict__ x,
    float* __restrict__ out,
    int V)
{
    const int v = (int)blockIdx.x * (int)blockDim.x + (int)threadIdx.x;
    if (v >= V) return;
    const int row    = (int)blockIdx.y;
    const int halfm1 = (V >> 1) - 1;
    const int j      = v > halfm1 ? halfm1 : v;
    const float c    = (4.0f * LOG2_10) * __builtin_amdgcn_rcpf((float)halfm1);
    const float f    = __builtin_exp2f(c * (float)j) * ONE_OVER_2PI;
    out[(size_t)row * (size_t)V + (size_t)v] = __builtin_amdgcn_sinf(x[row] * f);
}

extern "C" void kernel_launch(void* const* d_in, const int* in_sizes, int n_in,
                              void* d_out, int out_size, void* d_ws, size_t ws_size,
                              hipStream_t stream)
{
    const float* x = (const float*)d_in[0];
    float* out     = (float*)d_out;

    const int n_rows = in_sizes[0];            // 64*4096 = 262144
    const int V      = out_size / n_rows;      // 256

    const int ROWS_PER_BLOCK = 128;            // 128 KiB of output per block

    if (V == V256 && (n_rows % ROWS_PER_BLOCK) == 0) {
        const int grid = n_rows / ROWS_PER_BLOCK;                 // 2048
        sin_posenc256<<<grid, BLOCK, 0, stream>>>(x, out, n_rows, ROWS_PER_BLOCK);
    } else {
        dim3 grid((V + BLOCK - 1) / BLOCK, n_rows);
        sin_posenc_any<<<grid, BLOCK, 0, stream>>>(x, out, V);
    }
}